// MAIN_LAYER_48790828482887
// MI455X (gfx1250) — compile-verified
//
#include <hip/hip_runtime.h>

// ---------------------------------------------------------------------------
// 2-step staggered-grid DRP/FDTD solver, N=512, B=16, fp32.
// Bandwidth-bound (≈150MB compulsory traffic @ 23.3TB/s ≈ 6.5us); all interior
// DRP stencils are expressed as banded-matrix products executed on the matrix
// pipe with V_WMMA_F32_16X16X4_F32 (accumulating K-chunks of 4), boundary
// closures are scalar per-element overwrites. 192MB L2 absorbs stencil reuse.
// ---------------------------------------------------------------------------

typedef __attribute__((ext_vector_type(2))) float v2f;
typedef __attribute__((ext_vector_type(8))) float v8f;

#define NN   512
#define RDX  511.0f            // 1/DX
#define DTC  (1.0f / 1022.0f)  // DT = DX/2

// one-sided / centered closure coefficients (already * 1/DX)
#define CFA (RDX * (-23.0f/24.0f))
#define CFB (RDX * ( 21.0f/24.0f))
#define CFC (RDX * (  3.0f/24.0f))
#define CFD (RDX * ( -1.0f/24.0f))
#define CBA (RDX * (  1.0f/24.0f))
#define CBB (RDX * ( -3.0f/24.0f))
#define CBC (RDX * (-21.0f/24.0f))
#define CBD (RDX * ( 23.0f/24.0f))
#define C4A (RDX * (  1.0f/24.0f))
#define C4B (RDX * (-27.0f/24.0f))
#define C4C (RDX * ( 27.0f/24.0f))
#define C4D (RDX * ( -1.0f/24.0f))

__device__ __forceinline__ v8f wmma4(v2f a, v2f b, v8f c) {
  // D(16x16,f32) = A(16x4,f32) x B(4x16,f32) + C
  return __builtin_amdgcn_wmma_f32_16x16x4_f32(false, a, false, b, (short)0, c,
                                               false, false);
}

// band coefficient: Kc[d] if d in [0,3] else 0  (cndmask chain, no scratch)
__device__ __forceinline__ float bandc(float c0, float c1, float c2, float c3, int d) {
  float r = 0.0f;
  r = (d == 0) ? c0 : r;
  r = (d == 1) ? c1 : r;
  r = (d == 2) ? c2 : r;
  r = (d == 3) ? c3 : r;
  return r;
}

__device__ __forceinline__ float dot4(float a0, float a1, float a2, float a3,
                                      float x0, float x1, float x2, float x3) {
  return a0 * x0 + a1 * x1 + a2 * x2 + a3 * x3;
}

__device__ __forceinline__ int clampi(int v, int lo, int hi) {
  return v < lo ? lo : (v > hi ? hi : v);
}

// ===========================================================================
// amper: E_out = E + DT*(s1 - s2)
//   s1 = d/dh of Hy on E grid (interior via WMMA), s2 = d/dw of Hx on E grid.
// Each wave computes one 16x16 tile of the 512x512 E grid.
// ===========================================================================
__global__ __launch_bounds__(256)
void amper_kernel(const float* __restrict__ E, int e_bs,
                  const float* __restrict__ Hx, int hx_bs,
                  const float* __restrict__ Hy, int hy_bs,
                  const float* __restrict__ pbeta, const float* __restrict__ pdelta,
                  float* __restrict__ Eout, int eo_bs) {
  const int lane = threadIdx.x & 31;
  const int wid  = threadIdx.x >> 5;
  const int tile = blockIdx.x * 8 + wid;     // 1024 tiles = 32x32
  const int bb   = blockIdx.y;
  const int r0   = (tile >> 5) << 4;
  const int c0   = (tile & 31) << 4;

  const float beta  = pbeta[0];
  const float delta = pdelta[0];
  // combined DRP kernel rows (K[0]==K[2]==K0row, K[1]==K1row), * 1/DX
  const float K00 = 0.0f,         K01 = -beta * RDX;
  const float K02 = beta * RDX,   K03 = 0.0f;
  const float K10 = -delta * RDX;
  const float K11 = (2.0f * beta - 1.0f + 3.0f * delta) * RDX;
  const float K12 = (1.0f - 2.0f * beta - 3.0f * delta) * RDX;
  const float K13 = delta * RDX;

  const float* __restrict__ Eb  = E  + (long)bb * e_bs;   // 512 x 512
  const float* __restrict__ Hxb = Hx + (long)bb * hx_bs;  // 510 x 511
  const float* __restrict__ Hyb = Hy + (long)bb * hy_bs;  // 511 x 510
  float* __restrict__ Eo = Eout + (long)bb * eo_bs;

  const int mA = lane & 15;         // A-matrix M index == B-matrix N index
  const int kk = (lane >> 4) << 1;  // per-lane K pair within a 4-chunk

  v8f acc1 = {0.f, 0.f, 0.f, 0.f, 0.f, 0.f, 0.f, 0.f};  // s1
  v8f acc2 = {0.f, 0.f, 0.f, 0.f, 0.f, 0.f, 0.f, 0.f};  // s2

  // loop-invariant clamped coordinates
  const int cSL = clampi(c0 + mA - 2, 0, 509);  // Hy col w-2
  const int cSR = clampi(c0 + mA,     0, 509);  // Hy col w
  const int cSC = clampi(c0 + mA - 1, 0, 509);  // Hy col w-1
  const int rX2 = clampi(r0 + mA - 2, 0, 509);  // Hx row h-2
  const int rX1 = clampi(r0 + mA - 1, 0, 509);  // Hx row h-1
  const int rX0 = clampi(r0 + mA,     0, 509);  // Hx row h

#pragma unroll
  for (int k0 = 0; k0 < 20; k0 += 4) {
    const int kx = k0 + kk, ky = kx + 1;
    v2f band0, band1;  // Band[m][k] = K?[k-m]  (same values serve as BandT[k][n])
    band0.x = bandc(K00, K01, K02, K03, kx - mA);
    band0.y = bandc(K00, K01, K02, K03, ky - mA);
    band1.x = bandc(K10, K11, K12, K13, kx - mA);
    band1.y = bandc(K10, K11, K12, K13, ky - mA);

    // s1: Band(K0) x (Hy[.,w-2]+Hy[.,w]) + Band(K1) x Hy[.,w-1], rows r0-2+k
    const int ry0 = clampi(r0 - 2 + kx, 0, 510);
    const int ry1 = clampi(r0 - 2 + ky, 0, 510);
    v2f bsum, bc;
    bsum.x = Hyb[ry0 * 510 + cSL] + Hyb[ry0 * 510 + cSR];
    bsum.y = Hyb[ry1 * 510 + cSL] + Hyb[ry1 * 510 + cSR];
    bc.x   = Hyb[ry0 * 510 + cSC];
    bc.y   = Hyb[ry1 * 510 + cSC];
    acc1 = wmma4(band0, bsum, acc1);
    acc1 = wmma4(band1, bc,   acc1);

    // s2: (Hx[h-2,.]+Hx[h,.]) x BandT(K0) + Hx[h-1,.] x BandT(K1), cols c0-2+k
    const int cx = clampi(c0 - 2 + kx, 0, 510);
    const int cy = clampi(c0 - 2 + ky, 0, 510);
    v2f asum, ac;
    asum.x = Hxb[rX2 * 511 + cx] + Hxb[rX0 * 511 + cx];
    asum.y = Hxb[rX2 * 511 + cy] + Hxb[rX0 * 511 + cy];
    ac.x   = Hxb[rX1 * 511 + cx];
    ac.y   = Hxb[rX1 * 511 + cy];
    acc2 = wmma4(asum, band0, acc2);
    acc2 = wmma4(ac,   band1, acc2);
  }

  const int hi = lane >> 4;
  const int w  = c0 + mA;
#pragma unroll
  for (int e = 0; e < 8; ++e) {
    const int m = e + (hi << 3);
    const int h = r0 + m;
    float s1, s2;
    if (h >= 2 && h <= 509 && w >= 2 && w <= 509) {
      s1 = acc1[e];
      s2 = acc2[e];
    } else {
      s1 = 0.0f;
      s2 = 0.0f;
      if (h == 1 && w >= 1 && w <= 510)
        s1 = dot4(CFA, CFB, CFC, CFD, Hyb[w - 1], Hyb[510 + w - 1],
                  Hyb[2 * 510 + w - 1], Hyb[3 * 510 + w - 1]);
      else if (h == 510 && w >= 1 && w <= 510)
        s1 = dot4(CBA, CBB, CBC, CBD, Hyb[507 * 510 + w - 1], Hyb[508 * 510 + w - 1],
                  Hyb[509 * 510 + w - 1], Hyb[510 * 510 + w - 1]);
      else if ((w == 1 || w == 510) && h >= 2 && h <= 509) {
        const int wc = (w == 1) ? 0 : 509;
        s1 = dot4(C4A, C4B, C4C, C4D, Hyb[(h - 2) * 510 + wc], Hyb[(h - 1) * 510 + wc],
                  Hyb[h * 510 + wc], Hyb[(h + 1) * 510 + wc]);
      }
      if (w == 1 && h >= 1 && h <= 510)
        s2 = dot4(CFA, CFB, CFC, CFD, Hxb[(h - 1) * 511 + 0], Hxb[(h - 1) * 511 + 1],
                  Hxb[(h - 1) * 511 + 2], Hxb[(h - 1) * 511 + 3]);
      else if (w == 510 && h >= 1 && h <= 510)
        s2 = dot4(CBA, CBB, CBC, CBD, Hxb[(h - 1) * 511 + 507], Hxb[(h - 1) * 511 + 508],
                  Hxb[(h - 1) * 511 + 509], Hxb[(h - 1) * 511 + 510]);
      else if ((h == 1 || h == 510) && w >= 2 && w <= 509) {
        const int hr = (h == 1) ? 0 : 509;
        s2 = dot4(C4A, C4B, C4C, C4D, Hxb[hr * 511 + w - 2], Hxb[hr * 511 + w - 1],
                  Hxb[hr * 511 + w], Hxb[hr * 511 + w + 1]);
      }
    }
    Eo[h * NN + w] = Eb[h * NN + w] + DTC * (s1 - s2);
  }
}

// ===========================================================================
// faradayX: Hx_out[h][w] = Hx - DT*s3, s3 = d/dw of E at Hx points (510 x 511)
// ===========================================================================
__global__ __launch_bounds__(256)
void faradayx_kernel(const float* __restrict__ E, int e_bs,
                     const float* __restrict__ Hx, int hx_bs,
                     const float* __restrict__ pbeta, const float* __restrict__ pdelta,
                     float* __restrict__ Hxout, int ho_bs) {
  const int lane = threadIdx.x & 31;
  const int wid  = threadIdx.x >> 5;
  const int tile = blockIdx.x * 8 + wid;
  const int bb   = blockIdx.y;
  const int r0   = (tile >> 5) << 4;
  const int c0   = (tile & 31) << 4;

  const float beta  = pbeta[0];
  const float delta = pdelta[0];
  const float K00 = 0.0f,       K01 = -beta * RDX;
  const float K02 = beta * RDX, K03 = 0.0f;
  const float K10 = -delta * RDX;
  const float K11 = (2.0f * beta - 1.0f + 3.0f * delta) * RDX;
  const float K12 = (1.0f - 2.0f * beta - 3.0f * delta) * RDX;
  const float K13 = delta * RDX;

  const float* __restrict__ Eb  = E  + (long)bb * e_bs;
  const float* __restrict__ Hxb = Hx + (long)bb * hx_bs;
  float* __restrict__ Ho = Hxout + (long)bb * ho_bs;

  const int mA = lane & 15;
  const int kk = (lane >> 4) << 1;
  v8f acc = {0.f, 0.f, 0.f, 0.f, 0.f, 0.f, 0.f, 0.f};

  const int rE0 = clampi(r0 + mA,     0, 511);
  const int rE1 = clampi(r0 + mA + 1, 0, 511);
  const int rE2 = clampi(r0 + mA + 2, 0, 511);

#pragma unroll
  for (int k0 = 0; k0 < 20; k0 += 4) {
    const int kx = k0 + kk, ky = kx + 1;
    v2f band0, band1;
    band0.x = bandc(K00, K01, K02, K03, kx - mA);
    band0.y = bandc(K00, K01, K02, K03, ky - mA);
    band1.x = bandc(K10, K11, K12, K13, kx - mA);
    band1.y = bandc(K10, K11, K12, K13, ky - mA);
    // s3: (E[h,.]+E[h+2,.]) x BandT(K0) + E[h+1,.] x BandT(K1), cols c0-1+k
    const int cx = clampi(c0 - 1 + kx, 0, 511);
    const int cy = clampi(c0 - 1 + ky, 0, 511);
    v2f asum, ac;
    asum.x = Eb[rE0 * NN + cx] + Eb[rE2 * NN + cx];
    asum.y = Eb[rE0 * NN + cy] + Eb[rE2 * NN + cy];
    ac.x   = Eb[rE1 * NN + cx];
    ac.y   = Eb[rE1 * NN + cy];
    acc = wmma4(asum, band0, acc);
    acc = wmma4(ac,   band1, acc);
  }

  const int hi = lane >> 4;
  const int w  = c0 + mA;
#pragma unroll
  for (int e = 0; e < 8; ++e) {
    const int m = e + (hi << 3);
    const int h = r0 + m;
    if (h <= 509 && w <= 510) {
      float s3;
      if (w >= 1 && w <= 509)
        s3 = acc[e];
      else if (w == 0)
        s3 = dot4(CFA, CFB, CFC, CFD, Eb[(h + 1) * NN + 0], Eb[(h + 1) * NN + 1],
                  Eb[(h + 1) * NN + 2], Eb[(h + 1) * NN + 3]);
      else
        s3 = dot4(CBA, CBB, CBC, CBD, Eb[(h + 1) * NN + 508], Eb[(h + 1) * NN + 509],
                  Eb[(h + 1) * NN + 510], Eb[(h + 1) * NN + 511]);
      Ho[h * 511 + w] = Hxb[h * 511 + w] - DTC * s3;
    }
  }
}

// ===========================================================================
// faradayY: Hy_out[h][w] = Hy + DT*s4, s4 = d/dh of E at Hy points (511 x 510)
// ===========================================================================
__global__ __launch_bounds__(256)
void faradayy_kernel(const float* __restrict__ E, int e_bs,
                     const float* __restrict__ Hy, int hy_bs,
                     const float* __restrict__ pbeta, const float* __restrict__ pdelta,
                     float* __restrict__ Hyout, int ho_bs) {
  const int lane = threadIdx.x & 31;
  const int wid  = threadIdx.x >> 5;
  const int tile = blockIdx.x * 8 + wid;
  const int bb   = blockIdx.y;
  const int r0   = (tile >> 5) << 4;
  const int c0   = (tile & 31) << 4;

  const float beta  = pbeta[0];
  const float delta = pdelta[0];
  const float K00 = 0.0f,       K01 = -beta * RDX;
  const float K02 = beta * RDX, K03 = 0.0f;
  const float K10 = -delta * RDX;
  const float K11 = (2.0f * beta - 1.0f + 3.0f * delta) * RDX;
  const float K12 = (1.0f - 2.0f * beta - 3.0f * delta) * RDX;
  const float K13 = delta * RDX;

  const float* __restrict__ Eb  = E  + (long)bb * e_bs;
  const float* __restrict__ Hyb = Hy + (long)bb * hy_bs;
  float* __restrict__ Ho = Hyout + (long)bb * ho_bs;

  const int mA = lane & 15;
  const int kk = (lane >> 4) << 1;
  v8f acc = {0.f, 0.f, 0.f, 0.f, 0.f, 0.f, 0.f, 0.f};

  const int cB0 = clampi(c0 + mA,     0, 511);
  const int cB1 = clampi(c0 + mA + 1, 0, 511);
  const int cB2 = clampi(c0 + mA + 2, 0, 511);

#pragma unroll
  for (int k0 = 0; k0 < 20; k0 += 4) {
    const int kx = k0 + kk, ky = kx + 1;
    v2f band0, band1;
    band0.x = bandc(K00, K01, K02, K03, kx - mA);
    band0.y = bandc(K00, K01, K02, K03, ky - mA);
    band1.x = bandc(K10, K11, K12, K13, kx - mA);
    band1.y = bandc(K10, K11, K12, K13, ky - mA);
    // s4: Band(K0) x (E[.,w]+E[.,w+2]) + Band(K1) x E[.,w+1], rows r0-1+k
    const int ry0 = clampi(r0 - 1 + kx, 0, 511);
    const int ry1 = clampi(r0 - 1 + ky, 0, 511);
    v2f bsum, bc;
    bsum.x = Eb[ry0 * NN + cB0] + Eb[ry0 * NN + cB2];
    bsum.y = Eb[ry1 * NN + cB0] + Eb[ry1 * NN + cB2];
    bc.x   = Eb[ry0 * NN + cB1];
    bc.y   = Eb[ry1 * NN + cB1];
    acc = wmma4(band0, bsum, acc);
    acc = wmma4(band1, bc,   acc);
  }

  const int hi = lane >> 4;
  const int w  = c0 + mA;
#pragma unroll
  for (int e = 0; e < 8; ++e) {
    const int m = e + (hi << 3);
    const int h = r0 + m;
    if (h <= 510 && w <= 509) {
      float s4;
      if (h >= 1 && h <= 509)
        s4 = acc[e];
      else if (h == 0)
        s4 = dot4(CFA, CFB, CFC, CFD, Eb[0 * NN + w + 1], Eb[1 * NN + w + 1],
                  Eb[2 * NN + w + 1], Eb[3 * NN + w + 1]);
      else
        s4 = dot4(CBA, CBB, CBC, CBD, Eb[508 * NN + w + 1], Eb[509 * NN + w + 1],
                  Eb[510 * NN + w + 1], Eb[511 * NN + w + 1]);
      Ho[h * 510 + w] = Hyb[h * 510 + w] + DTC * s4;
    }
  }
}

// ===========================================================================
// Energy: deterministic two-stage weighted sum of squares (trapezoid weights)
// ===========================================================================
__global__ __launch_bounds__(256)
void energy_partial_kernel(const float* __restrict__ src, int batch_stride,
                           int H, int W, float* __restrict__ part) {
  __shared__ float red[256];
  const int bb = blockIdx.y;
  const int rb = blockIdx.x;  // 32 row-blocks of 16 rows
  const float* __restrict__ s = src + (long)bb * batch_stride;
  float acc = 0.0f;
  const int base = rb * 16;
#pragma unroll 1
  for (int hh = 0; hh < 16; ++hh) {
    const int h = base + hh;
    if (h >= H) break;
    const float ch = (h == 0 || h == H - 1) ? 0.5f : 1.0f;
    for (int w = threadIdx.x; w < W; w += 256) {
      const float cw = (w == 0 || w == W - 1) ? 0.5f : 1.0f;
      const float g = s[(long)h * W + w];
      acc += g * g * (ch * cw);
    }
  }
  red[threadIdx.x] = acc;
  __syncthreads();
  for (int off = 128; off > 0; off >>= 1) {
    if (threadIdx.x < (unsigned)off) red[threadIdx.x] += red[threadIdx.x + off];
    __syncthreads();
  }
  if (threadIdx.x == 0) part[bb * 32 + rb] = red[0];
}

__global__ void energy_final_kernel(const float* __restrict__ ws,
                                    float* __restrict__ out_inte,
                                    float* __restrict__ out_inth) {
  const int b = threadIdx.x;
  if (b >= 16) return;
  float se = 0.f, sx = 0.f, sy = 0.f;
  for (int i = 0; i < 32; ++i) {
    se += ws[b * 32 + i];
    sx += ws[512 + b * 32 + i];
    sy += ws[1024 + b * 32 + i];
  }
  const float DX2 = (1.0f / 511.0f) * (1.0f / 511.0f);
  out_inte[b] = DX2 * se;
  out_inth[b] = DX2 * (sx + sy);
}

// ===========================================================================
extern "C" void kernel_launch(void* const* d_in, const int* in_sizes, int n_in,
                              void* d_out, int out_size, void* d_ws, size_t ws_size,
                              hipStream_t stream) {
  (void)in_sizes; (void)n_in; (void)out_size; (void)ws_size;
  const float* E  = (const float*)d_in[0];  // [16,512,512]
  const float* Hx = (const float*)d_in[1];  // [16,510,511]
  const float* Hy = (const float*)d_in[2];  // [16,511,510]
  const float* pb = (const float*)d_in[3];  // beta scalar
  const float* pd = (const float*)d_in[4];  // delta scalar
  float* out = (float*)d_out;
  float* ws  = (float*)d_ws;

  const int  SE  = 1024 * 512;              // E-section batch stride
  const long OHX = 16L * SE;                // 8,388,608
  const int  SHX = 1020 * 511;              // 521,220
  const long OHY = OHX + 16L * SHX;         // 16,728,128
  const int  SHY = 1022 * 510;              // 521,220
  const long OI  = OHY + 16L * SHY;         // 25,067,648

  float* En  = out;                         // rows 0..511 of E-cat
  float* Em  = out + 512 * 512;             // rows 512..1023 of E-cat
  float* Hxn = out + OHX;
  float* Hxm = out + OHX + 510 * 511;
  float* Hyn = out + OHY;
  float* Hym = out + OHY + 511 * 510;

  dim3 grid(128, 16), blk(256);
  // ---- step 1 ----
  amper_kernel<<<grid, blk, 0, stream>>>(E, 512 * 512, Hx, 510 * 511,
                                         Hy, 511 * 510, pb, pd, En, SE);
  faradayx_kernel<<<grid, blk, 0, stream>>>(En, SE, Hx, 510 * 511, pb, pd, Hxn, SHX);
  faradayy_kernel<<<grid, blk, 0, stream>>>(En, SE, Hy, 511 * 510, pb, pd, Hyn, SHY);
  // ---- step 2 ----
  amper_kernel<<<grid, blk, 0, stream>>>(En, SE, Hxn, SHX, Hyn, SHY, pb, pd, Em, SE);
  faradayx_kernel<<<grid, blk, 0, stream>>>(Em, SE, Hxn, SHX, pb, pd, Hxm, SHX);
  faradayy_kernel<<<grid, blk, 0, stream>>>(Em, SE, Hyn, SHY, pb, pd, Hym, SHY);
  // ---- energies on step-1 fields ----
  dim3 eg(32, 16);
  energy_partial_kernel<<<eg, blk, 0, stream>>>(En, SE, 512, 512, ws);
  energy_partial_kernel<<<eg, blk, 0, stream>>>(Hxn, SHX, 510, 511, ws + 512);
  energy_partial_kernel<<<eg, blk, 0, stream>>>(Hyn, SHY, 511, 510, ws + 1024);
  energy_final_kernel<<<1, 32, 0, stream>>>(ws, out + OI, out + OI + 16);
}